// CDLoss_42932493091334
// MI455X (gfx1250) — compile-verified
//
#include <hip/hip_runtime.h>
#include <hip/hip_bf16.h>

typedef __attribute__((ext_vector_type(2))) float v2f;
typedef __attribute__((ext_vector_type(8))) float v8f;

#define NPTS  4096
#define NB    16         // b*s batches
#define TILES (NPTS/16)  // 256 column tiles

// One wave (32 threads) owns 32 rows of X (two 16-row WMMA tiles) for one
// (direction, batch) and scans all 4096 points of Y in 16-wide column tiles.
// Per tile-pair: ONE shared B build + TWO V_WMMA_F32_16X16X4_F32.
//
// Folding tricks (both exact):
//  * K=4 but d=3: spare K slot carries A[M][3]=1, B[3][N]=y2[N], so the WMMA
//    emits  D = -2*x.y + y2  directly with C = 0 (loop-invariant).
//  * x2[row] is constant under min over columns -> added once in epilogue.
//  * sqrt and max(.,0) are monotone -> applied after the min (131K sqrts
//    instead of 268M).
__global__ __launch_bounds__(32)
void chamfer_minsq_kernel(const float* __restrict__ pred,
                          const float* __restrict__ targ,
                          float* __restrict__ rowmin /* [2][NB][NPTS] */) {
  const int  lane = threadIdx.x;   // 0..31
  const int  rblk = blockIdx.x;    // 0..127 : 32-row block
  const int  m    = blockIdx.y;    // 0..15  : batch
  const int  dir  = blockIdx.z;    // 0: X=pred,Y=targ (dr) ; 1: swapped (dl)

  const float* X = (dir == 0 ? pred : targ) + (size_t)m * NPTS * 3;
  const float* Y = (dir == 0 ? targ : pred) + (size_t)m * NPTS * 3;

  const int  rlo = lane & 15;
  const bool hi  = lane >= 16;

  // Two 16-row A tiles: rows r0 = rblk*32 + rlo and r1 = r0 + 16.
  const int r0 = rblk * 32 + rlo;
  const int r1 = r0 + 16;
  const float p0x = X[r0 * 3 + 0], p0y = X[r0 * 3 + 1], p0z = X[r0 * 3 + 2];
  const float p1x = X[r1 * 3 + 0], p1y = X[r1 * 3 + 1], p1z = X[r1 * 3 + 2];
  const float x2_0 = p0x * p0x + p0y * p0y + p0z * p0z;
  const float x2_1 = p1x * p1x + p1y * p1y + p1z * p1z;

  // A-matrix 16x4 f32 layout: VGPR0 = K0 (lanes 0-15) | K2 (lanes 16-31);
  // VGPR1 = K1 | K3.  Data pre-scaled by -2; K3 slot = 1.0 (y2 fold).
  v2f a0, a1;
  a0.x = -2.0f * (hi ? p0z : p0x);
  a0.y = hi ? 1.0f : (-2.0f * p0y);
  a1.x = -2.0f * (hi ? p1z : p1x);
  a1.y = hi ? 1.0f : (-2.0f * p1y);

  v8f zc = {};  // C = 0, loop-invariant
  v8f rmin0, rmin1;
#pragma unroll
  for (int v = 0; v < 8; ++v) { rmin0[v] = 3.0e38f; rmin1[v] = 3.0e38f; }

  for (int j = 0; j < TILES; ++j) {
    const int ycol = j * 16 + rlo;
    const float tx = Y[ycol * 3 + 0];
    const float ty = Y[ycol * 3 + 1];
    const float tz = Y[ycol * 3 + 2];
    const float y2 = tx * tx + ty * ty + tz * tz;

    // B-matrix 4x16: B[k][n] = Y[n][k] for k<3, B[3][n] = y2[n].
    // Lane = column n; VGPR0 = K0 | K2, VGPR1 = K1 | K3.
    v2f bm;
    bm.x = hi ? tz : tx;
    bm.y = hi ? y2 : ty;

    // D = (-2X)*Y^T + y2  (squared distance minus the row-constant x2)
    v8f d0 = __builtin_amdgcn_wmma_f32_16x16x4_f32(
        false, a0, false, bm, (short)0, zc, false, false);
    v8f d1 = __builtin_amdgcn_wmma_f32_16x16x4_f32(
        false, a1, false, bm, (short)0, zc, false, false);

#pragma unroll
    for (int v = 0; v < 8; ++v) {
      rmin0[v] = fminf(rmin0[v], d0[v]);
      rmin1[v] = fminf(rmin1[v], d1[v]);
    }
  }

  // Min across the 16 columns: xor masks < 16 reduce each 16-lane half
  // independently (wave32).
#pragma unroll
  for (int v = 0; v < 8; ++v) {
    float t0 = rmin0[v], t1 = rmin1[v];
    t0 = fminf(t0, __shfl_xor(t0, 1, 32));
    t1 = fminf(t1, __shfl_xor(t1, 1, 32));
    t0 = fminf(t0, __shfl_xor(t0, 2, 32));
    t1 = fminf(t1, __shfl_xor(t1, 2, 32));
    t0 = fminf(t0, __shfl_xor(t0, 4, 32));
    t1 = fminf(t1, __shfl_xor(t1, 4, 32));
    t0 = fminf(t0, __shfl_xor(t0, 8, 32));
    t1 = fminf(t1, __shfl_xor(t1, 8, 32));
    rmin0[v] = t0;
    rmin1[v] = t1;
  }

  // rx2_*[v] = x2 of the row that (VGPR v, this lane half) maps to:
  // C/D element (VGPR v, lane l) is row M = v + 8*(l>=16); x2[M] is in lane M.
  float rx2_0[8], rx2_1[8];
#pragma unroll
  for (int v = 0; v < 8; ++v) {
    const int src = v + (hi ? 8 : 0);
    rx2_0[v] = __shfl(x2_0, src, 32);
    rx2_1[v] = __shfl(x2_1, src, 32);
  }

  // Lane 0 writes rows M=0..7, lane 16 writes rows M=8..15 of each tile.
  if (rlo == 0) {
    float* base = rowmin + ((size_t)dir * NB + m) * NPTS + rblk * 32 + (hi ? 8 : 0);
#pragma unroll
    for (int v = 0; v < 8; ++v) {
      base[v]      = sqrtf(fmaxf(rmin0[v] + rx2_0[v], 0.0f));
      base[16 + v] = sqrtf(fmaxf(rmin1[v] + rx2_1[v], 0.0f));
    }
  }
}

// Deterministic single-workgroup tree reduction of the 2*16*4096 row-mins.
// loss = mean_batch( mean dl + mean dr ) = (sum of all entries) / 65536.
__global__ __launch_bounds__(1024)
void chamfer_reduce_kernel(const float* __restrict__ rowmin,
                           float* __restrict__ out) {
  __shared__ float sm[1024];
  const int tid = threadIdx.x;
  const int TOT = 2 * NB * NPTS;  // 131072
  float s = 0.0f;
  for (int i = tid; i < TOT; i += 1024) s += rowmin[i];
  sm[tid] = s;
  __syncthreads();
  for (int off = 512; off > 0; off >>= 1) {
    if (tid < off) sm[tid] += sm[tid + off];
    __syncthreads();
  }
  if (tid == 0) out[0] = sm[0] * (1.0f / 65536.0f);
}

extern "C" void kernel_launch(void* const* d_in, const int* in_sizes, int n_in,
                              void* d_out, int out_size, void* d_ws, size_t ws_size,
                              hipStream_t stream) {
  const float* pred = (const float*)d_in[0];
  const float* targ = (const float*)d_in[1];
  float* rowmin = (float*)d_ws;   // 2*16*4096 floats = 512 KB scratch

  dim3 grid(TILES / 2, NB, 2);    // 4096 waves of 32 threads, 32 rows each
  chamfer_minsq_kernel<<<grid, dim3(32), 0, stream>>>(pred, targ, rowmin);
  chamfer_reduce_kernel<<<1, dim3(1024), 0, stream>>>(rowmin, (float*)d_out);
}